// MaskInfluencerLoss_12841952215057
// MI455X (gfx1250) — compile-verified
//
#include <hip/hip_runtime.h>
#include <hip/hip_bf16.h>

typedef __attribute__((ext_vector_type(2))) float v2f;
typedef __attribute__((ext_vector_type(4))) float v4f;
typedef __attribute__((ext_vector_type(8))) float v8f;

#define NCOLS 256
#define KSEG  200            // clusters 1..200; label 0 = background
#define NLAB  201            // K+1
#define KPAD  208            // 13 * 16
#define NTILE 13

// workspace layout (floats)
#define OFF_SEG   0                         // 201*256   segment sums of logp
#define OFF_CNT   (OFF_SEG + NLAB*NCOLS)    // 256       label counts (201 used)
#define OFF_PROF  (OFF_CNT + 256)           // 208*256   profiles, zero padded
#define OFF_G     (OFF_PROF + KPAD*NCOLS)   // 208*208   Gram matrix
#define OFF_SCAL  (OFF_G + KPAD*KPAD)       // 16        [attr_sum, n_present, bg_sum, rep_sum, n_pairs]
#define OFF_PRES  (OFF_SCAL + 16)           // 208       present flags
#define WS_FLOATS (OFF_PRES + KPAD)

__global__ void zero_ws_kernel(float* ws, int n) {
    int i = blockIdx.x * blockDim.x + threadIdx.x;
    if (i < n) ws[i] = 0.0f;
}

// Pass 1: single nontemporal b128-per-lane stream over logits (bandwidth-dominant,
// ~205MB). Block-private 201*256 f32 segment table in LDS (201KB of the WGP's
// 320KB). 64 lanes cover one row (4 cols each); block walks 4 rows/iteration.
// Cross-row collisions on (label,col) resolved with native ds_add_f32 atomics.
__global__ void pass1_segsum(const float* __restrict__ logits,
                             const int* __restrict__ labels,
                             float* __restrict__ seg_sum,
                             float* __restrict__ counts,
                             float* __restrict__ scal,
                             int n_rows, int rows_per_block) {
    extern __shared__ float smem[];
    float* seg  = smem;                 // NLAB*NCOLS
    float* cntl = smem + NLAB*NCOLS;    // NLAB
    const int t = threadIdx.x;          // 0..255 (8 waves)

    for (int i = t; i < NLAB*NCOLS; i += NCOLS) seg[i] = 0.0f;
    if (t < NLAB) cntl[t] = 0.0f;
    __syncthreads();

    const int rgrp = t >> 6;            // wave-uniform: which of 4 rows
    const int c4   = (t & 63) * 4;      // 16B-aligned column base

    int r0 = blockIdx.x * rows_per_block;
    int r1 = r0 + rows_per_block; if (r1 > n_rows) r1 = n_rows;

    float bg = 0.0f;
    for (int r = r0 + rgrp; r < r1; r += 4) {
        int lab = labels[r];                                  // wave-uniform -> s_load
        const float* src = logits + (long)r * NCOLS + c4;
        __builtin_prefetch(src + 4 * NCOLS, 0, 0);            // next row group
        v4f x = __builtin_nontemporal_load((const v4f*)src);  // read-once stream: TH=NT

        v4f p, lp;
#pragma unroll
        for (int e = 0; e < 4; ++e) {
            p[e]  = 1.0f / (1.0f + __expf(-x[e]));            // sigmoid
            lp[e] = __logf(p[e] + 1e-6f);
        }
        float* dst = &seg[lab * NCOLS + c4];
#pragma unroll
        for (int e = 0; e < 4; ++e) unsafeAtomicAdd(dst + e, lp[e]);  // ds_add_f32
        if (lab == 0) bg += p[0] + p[1] + p[2] + p[3];
        if ((t & 63) == 0) unsafeAtomicAdd(&cntl[lab], 1.0f);
    }
    __syncthreads();

    for (int i = t; i < NLAB*NCOLS; i += NCOLS) {
        float v = seg[i];
        if (v != 0.0f) unsafeAtomicAdd(&seg_sum[i], v);       // global_atomic_add_f32
    }
    if (t < NLAB) {
        float v = cntl[t];
        if (v != 0.0f) unsafeAtomicAdd(&counts[t], v);
    }
    // wave32 reduction of background prob sum
    for (int off = 16; off; off >>= 1) bg += __shfl_down(bg, off, 32);
    if ((t & 31) == 0 && bg != 0.0f) unsafeAtomicAdd(&scal[2], bg);
}

// Pass 2: per-cluster log-mean, logsumexp (attractive), profiles = exp(log_gm).
__global__ void pass2_cluster(const float* __restrict__ seg_sum,
                              const float* __restrict__ counts,
                              float* __restrict__ prof,
                              float* __restrict__ present,
                              float* __restrict__ scal) {
    __shared__ float red[NCOLS];
    const int k = blockIdx.x;       // 0..199 -> label k+1
    const int m = threadIdx.x;

    float cnt = counts[k + 1];
    bool pres = cnt > 0.0f;
    float lg = seg_sum[(k + 1) * NCOLS + m] / fmaxf(cnt, 1.0f);
    prof[k * NCOLS + m] = pres ? expf(lg) : 0.0f;

    red[m] = lg; __syncthreads();
    for (int s = 128; s; s >>= 1) { if (m < s) red[m] = fmaxf(red[m], red[m + s]); __syncthreads(); }
    float mx = red[0]; __syncthreads();
    red[m] = expf(lg - mx); __syncthreads();
    for (int s = 128; s; s >>= 1) { if (m < s) red[m] += red[m + s]; __syncthreads(); }

    if (m == 0) {
        present[k] = pres ? 1.0f : 0.0f;
        if (pres) {
            float attr_k = -(mx + logf(red[0]));   // TEMP = 1
            unsafeAtomicAdd(&scal[0], attr_k);
            unsafeAtomicAdd(&scal[1], 1.0f);
        }
    }
}

// Pass 3: Gram matrix G = P * P^T via V_WMMA_F32_16X16X4_F32.
// One wave per 16x16 tile; 64 WMMA steps over the 256-wide K dim.
__global__ void pass3_gram_wmma(const float* __restrict__ P, float* __restrict__ G) {
    const int tile = blockIdx.x;          // 0..168
    const int ti = tile / NTILE, tj = tile % NTILE;
    const int lane = threadIdx.x;         // 0..31
    const int half = lane >> 4, ln = lane & 15;

    const float* arow = P + (long)(ti * 16 + ln) * NCOLS;
    const float* brow = P + (long)(tj * 16 + ln) * NCOLS;   // symmetric operand

    v8f c = {};
    for (int kk = 0; kk < NCOLS; kk += 4) {
        v2f a, b;
        a.x = arow[kk + half * 2 + 0];
        a.y = arow[kk + half * 2 + 1];
        b.x = brow[kk + half * 2 + 0];
        b.y = brow[kk + half * 2 + 1];
        c = __builtin_amdgcn_wmma_f32_16x16x4_f32(false, a, false, b,
                                                  (short)0, c, false, false);
    }
    // D layout: VGPR g -> rows g (lanes 0-15) / g+8 (lanes 16-31), col = lane%16
    const int col = tj * 16 + ln;
#pragma unroll
    for (int g = 0; g < 8; ++g) {
        int row = ti * 16 + g + half * 8;
        G[row * KPAD + col] = c[g];
    }
}

// Pass 4: hinge on pairwise distances from Gram diagonal identity.
__global__ void pass4_repulsive(const float* __restrict__ G,
                                const float* __restrict__ present,
                                float* __restrict__ scal) {
    int idx = blockIdx.x * blockDim.x + threadIdx.x;
    float h = 0.0f, np = 0.0f;
    if (idx < KSEG * KSEG) {
        int i = idx / KSEG, j = idx % KSEG;
        if (i != j && present[i] > 0.0f && present[j] > 0.0f) {
            float sq = G[i * KPAD + i] + G[j * KPAD + j] - 2.0f * G[i * KPAD + j];
            float dist = sqrtf(fmaxf(sq, 0.0f));
            h = fmaxf(1.0f - dist, 0.0f);     // MARGIN = 1
            np = 1.0f;
        }
    }
    for (int off = 16; off; off >>= 1) {
        h  += __shfl_down(h, off, 32);
        np += __shfl_down(np, off, 32);
    }
    if ((threadIdx.x & 31) == 0) {
        if (h  != 0.0f) unsafeAtomicAdd(&scal[3], h);
        if (np != 0.0f) unsafeAtomicAdd(&scal[4], np);
    }
}

__global__ void pass5_final(const float* __restrict__ scal,
                            const float* __restrict__ counts,
                            float* __restrict__ out) {
    if (threadIdx.x == 0 && blockIdx.x == 0) {
        float attractive = scal[0] / fmaxf(scal[1], 1.0f);
        float repulsive  = scal[3] / fmaxf(scal[4], 1.0f);
        float bg_loss    = scal[2] / (fmaxf(counts[0], 1.0f) * (float)NCOLS);
        out[0] = attractive + repulsive + bg_loss;
        out[1] = attractive;
        out[2] = repulsive;
        out[3] = bg_loss;
    }
}

extern "C" void kernel_launch(void* const* d_in, const int* in_sizes, int n_in,
                              void* d_out, int out_size, void* d_ws, size_t ws_size,
                              hipStream_t stream) {
    const float* logits = (const float*)d_in[0];
    const int*   labels = (const int*)d_in[1];
    float* out = (float*)d_out;
    float* ws  = (float*)d_ws;

    const int n_rows = in_sizes[1];                 // N = 200000
    const int grid1  = 200;
    const int rpb    = ((n_rows + grid1 - 1) / grid1 + 3) & ~3;  // multiple of 4
    const size_t lds_bytes = (size_t)(NLAB * NCOLS + NLAB) * sizeof(float);

    // Opt in to >64KB dynamic LDS (CDNA5 WGP has 320KB). Host-side, not stream-ordered.
    hipFuncSetAttribute((const void*)pass1_segsum,
                        hipFuncAttributeMaxDynamicSharedMemorySize, (int)lds_bytes);

    zero_ws_kernel<<<(WS_FLOATS + 255) / 256, 256, 0, stream>>>(ws, WS_FLOATS);

    pass1_segsum<<<grid1, NCOLS, lds_bytes, stream>>>(
        logits, labels, ws + OFF_SEG, ws + OFF_CNT, ws + OFF_SCAL, n_rows, rpb);

    pass2_cluster<<<KSEG, NCOLS, 0, stream>>>(
        ws + OFF_SEG, ws + OFF_CNT, ws + OFF_PROF, ws + OFF_PRES, ws + OFF_SCAL);

    pass3_gram_wmma<<<NTILE * NTILE, 32, 0, stream>>>(ws + OFF_PROF, ws + OFF_G);

    pass4_repulsive<<<(KSEG * KSEG + 255) / 256, 256, 0, stream>>>(
        ws + OFF_G, ws + OFF_PRES, ws + OFF_SCAL);

    pass5_final<<<1, 32, 0, stream>>>(ws + OFF_SCAL, ws + OFF_CNT, out);
}